// ProgramTransformer_6296422056700
// MI455X (gfx1250) — compile-verified
//
#include <hip/hip_runtime.h>
#include <stdint.h>

// ProgramTransformer sampling head for gfx1250 (MI455X).
// One wave32 per batch row; async global->LDS staging (ASYNCcnt path),
// threefry2x32 Gumbel-argmax categorical sampling, shuffle reductions.

#define NO 7
#define NT 64
#define NP 4
#define NL 7

struct Keys { uint32_t k[12]; };   // 6 threefry subkeys (hi,lo pairs)

__host__ __device__ inline void tf2x32(uint32_t k0, uint32_t k1,
                                       uint32_t c0, uint32_t c1,
                                       uint32_t& o0, uint32_t& o1) {
  const uint32_t ks2 = k0 ^ k1 ^ 0x1BD11BDAu;
  uint32_t x0 = c0 + k0, x1 = c1 + k1;
#define TF_R(r) { x0 += x1; x1 = (x1 << (r)) | (x1 >> (32 - (r))); x1 ^= x0; }
  TF_R(13) TF_R(15) TF_R(26) TF_R(6)  x0 += k1;  x1 += ks2 + 1u;
  TF_R(17) TF_R(29) TF_R(16) TF_R(24) x0 += ks2; x1 += k0 + 2u;
  TF_R(13) TF_R(15) TF_R(26) TF_R(6)  x0 += k0;  x1 += k1 + 3u;
  TF_R(17) TF_R(29) TF_R(16) TF_R(24) x0 += k1;  x1 += ks2 + 4u;
  TF_R(13) TF_R(15) TF_R(26) TF_R(6)  x0 += ks2; x1 += k0 + 5u;
#undef TF_R
  o0 = x0; o1 = x1;
}

// JAX gumbel noise for flat element f of an n-element draw (n even, half=n/2).
__device__ inline float jax_gumbel(uint32_t k0, uint32_t k1, uint32_t f, uint32_t half) {
  uint32_t o0, o1, bits;
  if (f < half) { tf2x32(k0, k1, f, half + f, o0, o1); bits = o0; }
  else          { tf2x32(k0, k1, f - half, f, o0, o1); bits = o1; }
  float u = __uint_as_float(0x3f800000u | (bits >> 9)) - 1.0f;  // [0,1)
  u = fmaxf(u, 1.1754943508222875e-38f);                        // minval=tiny
  return -logf(-logf(u));
}

// ---- wave32 butterfly reductions ----
__device__ inline float wmaxf(float v) {
#pragma unroll
  for (int o = 16; o > 0; o >>= 1) v = fmaxf(v, __shfl_xor(v, o, 32));
  return v;
}
__device__ inline float wsumf(float v) {
#pragma unroll
  for (int o = 16; o > 0; o >>= 1) v += __shfl_xor(v, o, 32);
  return v;
}
__device__ inline void wargmax(float& v, int& i) {   // first-occurrence ties
#pragma unroll
  for (int o = 16; o > 0; o >>= 1) {
    float ov = __shfl_xor(v, o, 32);
    int   oi = __shfl_xor(i, o, 32);
    if (ov > v || (ov == v && oi < i)) { v = ov; i = oi; }
  }
}

// ---- CDNA5 async global -> LDS (tracked with ASYNCcnt) ----
__device__ inline void async_g2l_b128(uint32_t lds_byte, uint64_t gaddr) {
  asm volatile("global_load_async_to_lds_b128 %0, %1, off"
               :: "v"(lds_byte), "v"(gaddr) : "memory");
}
__device__ inline void wait_async0() {
  asm volatile("s_wait_asynccnt 0" ::: "memory");
}

// Categorical over a 64-long row held 2 elems/lane (cols lane, lane+32).
__device__ inline void sample64(float s0, float s1, float m0, float m1,
                                float g0, float g1, int lane,
                                int& action, float& lp, float& lpm) {
  float v0 = s0 + g0, v1 = s1 + g1;
  float v; int idx;
  if (v1 > v0) { v = v1; idx = lane + 32; } else { v = v0; idx = lane; }
  wargmax(v, idx);
  action = idx;                                   // wave-uniform now
  float m = wmaxf(fmaxf(s0, s1));
  float e = expf(s0 - m) + expf(s1 - m);          // exp(-inf - m) = 0
  float lse = m + logf(wsumf(e));
  float sat = (action < 32) ? s0 : s1;            // uniform slot select
  sat = __shfl(sat, action & 31, 32);
  lp = sat - lse;
  int cnt = __popcll(__ballot(m0 == 0.0f)) + __popcll(__ballot(m1 == 0.0f));
  lpm = (cnt > 1) ? 1.0f : 0.0f;
}

__global__ __launch_bounds__(256) void pt_sample_kernel(
    const float* __restrict__ order_mask,
    const int*   __restrict__ tile_remain_budgets,
    const float* __restrict__ tile_masks,
    const int*   __restrict__ loop_ind_p,
    const float* __restrict__ remain_buffer_size,
    const int*   __restrict__ tile2_max_in,
    const float* __restrict__ max_temporal_tile2,
    const int*   __restrict__ sp_tile2_max_in,
    const int*   __restrict__ sp_tile2_min_in,
    const float* __restrict__ order_logit,
    const float* __restrict__ tile_logits,
    const float* __restrict__ sp_tile2_logit,
    float* __restrict__ out, int B, Keys K)
{
  __shared__ float smem[8 * 576];                 // 576 f32 per wave, 18.4 KB/WG
  const int lane = threadIdx.x & 31;
  const int wid  = threadIdx.x >> 5;
  const int b    = blockIdx.x * 8 + wid;
  if (b >= B) return;
  const int li = loop_ind_p[0];

  // ---- stage this row's streaming data into LDS via async copies ----
  // layout (floats): [0..255]=tile_logits[b,0..3,:]  [256..511]=tile_masks[b,li,0..3,:]
  //                  [512..575]=sp_tile2_logit[b,:]
  float* ws = &smem[wid * 576];
  uint32_t lds0 = (uint32_t)(uintptr_t)(void*)ws;           // LDS byte offset
  uint64_t gtl = (uint64_t)(uintptr_t)(const void*)(tile_logits + (size_t)b * (NP * NT));
  uint64_t gtm = (uint64_t)(uintptr_t)(const void*)(tile_masks + ((size_t)b * NL + li) * (NP * NT));
  uint64_t gsp = (uint64_t)(uintptr_t)(const void*)(sp_tile2_logit + (size_t)b * NT);
  async_g2l_b128(lds0 + lane * 16,                gtl + lane * 16);
  async_g2l_b128(lds0 + (lane + 32) * 16,         gtl + (size_t)(lane + 32) * 16);
  async_g2l_b128(lds0 + 1024 + lane * 16,         gtm + lane * 16);
  async_g2l_b128(lds0 + 1024 + (lane + 32) * 16,  gtm + (size_t)(lane + 32) * 16);
  if (lane < 16)
    async_g2l_b128(lds0 + 2048 + lane * 16,       gsp + lane * 16);

  // speculative prefetch of a later row's logits (global_prefetch_b8)
  __builtin_prefetch(tile_logits + ((size_t)b + 8) * (NP * NT), 0, 1);

  // ---- wave-uniform scalars (L2-resident, tiny) ----
  float rbs   = remain_buffer_size[b];
  int   t2cap = tile2_max_in[b];
  int   mtt2i = (int)max_temporal_tile2[b];       // astype(int32) truncation
  int   spmx  = sp_tile2_max_in[b];
  int   spmn  = sp_tile2_min_in[b];
  int   bud[3];
#pragma unroll
  for (int p = 1; p < NP; ++p)
    bud[p - 1] = tile_remain_budgets[((size_t)b * NL + li) * NP + p];

  const int t0 = lane, t1 = lane + 32;
  const uint32_t halfT = (uint32_t)B * (NT / 2);
  const uint32_t halfO = (uint32_t)B * NO / 2u;

  // ---- order sampling (7-wide) — overlaps the async-copy latency ----
  float oscore = -INFINITY, omaskv = 1.0f;
  if (lane < NO) {
    omaskv = order_mask[(size_t)b * NO + lane];
    oscore = order_logit[(size_t)b * NO + lane] + omaskv;
  }
  float ov = -INFINITY;
  if (lane < NO) ov = oscore + jax_gumbel(K.k[0], K.k[1], (uint32_t)b * NO + lane, halfO);
  int oi = lane;
  wargmax(ov, oi);
  const int order_action = oi;
  float om  = wmaxf(oscore);
  float oe  = (lane < NO) ? expf(oscore - om) : 0.0f;
  float olse = om + logf(wsumf(oe));
  float order_lp  = __shfl(oscore, order_action, 32) - olse;
  int   ozc = __popcll(__ballot(lane < NO && omaskv == 0.0f));
  float order_lpm = (ozc > 1) ? 1.0f : 0.0f;

  // ---- drain async copies, then read staged data from LDS ----
  wait_async0();
  volatile const float* wsv = ws;
  float tl00  = wsv[t0],        tl01  = wsv[t1];          // tile_logits p=0
  float base0 = wsv[256 + t0],  base1 = wsv[256 + t1];    // tile2_base
  float sl0   = wsv[512 + t0],  sl1   = wsv[512 + t1];    // sp logits

  // ---- sp_tile2 sampling ----
  float spm0 = (t0 > spmx || t0 < spmn) ? -INFINITY : base0;
  float spm1 = (t1 > spmx || t1 < spmn) ? -INFINITY : base1;
  float spg0 = jax_gumbel(K.k[2], K.k[3], (uint32_t)b * NT + t0, halfT);
  float spg1 = jax_gumbel(K.k[2], K.k[3], (uint32_t)b * NT + t1, halfT);
  int sp_action; float sp_lp, sp_lpm;
  sample64(sl0 + spm0, sl1 + spm1, spm0, spm1, spg0, spg1, lane, sp_action, sp_lp, sp_lpm);

  // ---- tile2 sampling ----
  int t2min = sp_action;
  int t2hi  = t2min + mtt2i; if (t2cap < t2hi) t2hi = t2cap;
  float t2m0 = (t0 > t2hi || t0 < t2min) ? -INFINITY : base0;
  float t2m1 = (t1 > t2hi || t1 < t2min) ? -INFINITY : base1;
  float t2g0 = jax_gumbel(K.k[4], K.k[5], (uint32_t)b * NT + t0, halfT);
  float t2g1 = jax_gumbel(K.k[4], K.k[5], (uint32_t)b * NT + t1, halfT);
  int t2_action; float t2_lp, t2_lpm;
  sample64(tl00 + t2m0, tl01 + t2m1, t2m0, t2m1, t2g0, t2g1, lane, t2_action, t2_lp, t2_lpm);

  // ---- serial p = 1..3 chain (rbs depends on previous action) ----
  const float primes[3] = {2.0f, 3.0f, 5.0f};
  const float lg2p[3]   = {1.5849625007211562f,  // log2(3)
                           2.3219280948873622f,  // log2(5)
                           2.8073549220576042f}; // log2(7)
  int   acts[3]; float lps[3], lpms[3];
  int tile_action = t2_action;
#pragma unroll
  for (int p = 1; p < NP; ++p) {
    rbs = rbs / powf(primes[p - 1], (float)tile_action);
    int tmax = (int)(log2f(fmaxf(rbs, 1.0f)) / lg2p[p - 1]);
    if (tmax < 0) tmax = 0;
    if (tmax > NT - 1) tmax = NT - 1;
    if (tmax > bud[p - 1]) tmax = bud[p - 1];
    float mk0 = (t0 > tmax) ? -INFINITY : wsv[256 + p * NT + t0];
    float mk1 = (t1 > tmax) ? -INFINITY : wsv[256 + p * NT + t1];
    float sc0 = wsv[p * NT + t0] + mk0;
    float sc1 = wsv[p * NT + t1] + mk1;
    float gg0 = jax_gumbel(K.k[2 * (2 + p)], K.k[2 * (2 + p) + 1], (uint32_t)b * NT + t0, halfT);
    float gg1 = jax_gumbel(K.k[2 * (2 + p)], K.k[2 * (2 + p) + 1], (uint32_t)b * NT + t1, halfT);
    int a; float lp, lpm;
    sample64(sc0, sc1, mk0, mk1, gg0, gg1, lane, a, lp, lpm);
    acts[p - 1] = a; lps[p - 1] = lp; lpms[p - 1] = lpm;
    tile_action = a;
  }

  // ---- write 21 outputs per row (all wave-uniform) ----
  if (lane == 0) {
    out[b] = (float)order_action;
    float* ta  = out + (size_t)B;           // (B,4) tile actions
    float* sta = out + (size_t)B * 5;       // (B,4) sp tile actions
    float* lpo = out + (size_t)B * 9;       // (B,6) log probs
    float* lmo = out + (size_t)B * 15;      // (B,6) log prob masks
    ta[(size_t)b * 4 + 0] = (float)t2_action;
    ta[(size_t)b * 4 + 1] = (float)acts[0];
    ta[(size_t)b * 4 + 2] = (float)acts[1];
    ta[(size_t)b * 4 + 3] = (float)acts[2];
    sta[(size_t)b * 4 + 0] = (float)sp_action;
    sta[(size_t)b * 4 + 1] = 0.0f;
    sta[(size_t)b * 4 + 2] = 0.0f;
    sta[(size_t)b * 4 + 3] = 0.0f;
    lpo[(size_t)b * 6 + 0] = order_lp;
    lpo[(size_t)b * 6 + 1] = t2_lp;
    lpo[(size_t)b * 6 + 2] = lps[0];
    lpo[(size_t)b * 6 + 3] = lps[1];
    lpo[(size_t)b * 6 + 4] = lps[2];
    lpo[(size_t)b * 6 + 5] = sp_lp;
    lmo[(size_t)b * 6 + 0] = order_lpm;
    lmo[(size_t)b * 6 + 1] = t2_lpm;
    lmo[(size_t)b * 6 + 2] = lpms[0];
    lmo[(size_t)b * 6 + 3] = lpms[1];
    lmo[(size_t)b * 6 + 4] = lpms[2];
    lmo[(size_t)b * 6 + 5] = sp_lpm;
  }
}

extern "C" void kernel_launch(void* const* d_in, const int* in_sizes, int n_in,
                              void* d_out, int out_size, void* d_ws, size_t ws_size,
                              hipStream_t stream) {
  (void)n_in; (void)out_size; (void)d_ws; (void)ws_size;
  // setup_inputs order:
  // 0 program_seq 1 order_mask 2 tile_remain_budgets 3 tile_masks 4 cur_buffer_level
  // 5 loop_ind 6 remain_buffer_size 7 tile2_max 8 max_temporal_tile2
  // 9 sp_tile2_max 10 sp_tile2_min 11 order_logit 12 tile_logits 13 sp_tile2_logit
  const float* order_mask = (const float*)d_in[1];
  const int*   budgets    = (const int*)d_in[2];
  const float* tile_masks = (const float*)d_in[3];
  const int*   loop_ind   = (const int*)d_in[5];
  const float* rbs        = (const float*)d_in[6];
  const int*   t2max      = (const int*)d_in[7];
  const float* mtt2       = (const float*)d_in[8];
  const int*   spmax      = (const int*)d_in[9];
  const int*   spmin      = (const int*)d_in[10];
  const float* olog       = (const float*)d_in[11];
  const float* tlog       = (const float*)d_in[12];
  const float* splog      = (const float*)d_in[13];
  float* out = (float*)d_out;
  const int B = in_sizes[6];

  // keys = jax.random.split(key(1), 6): counts = iota(12), halves -> out pairs
  Keys K;
  uint32_t y0[6], y1[6];
  for (int i = 0; i < 6; ++i)
    tf2x32(0u, 1u, (uint32_t)i, (uint32_t)(6 + i), y0[i], y1[i]);
  for (int i = 0; i < 3; ++i) {
    K.k[2 * i]         = y0[2 * i];
    K.k[2 * i + 1]     = y0[2 * i + 1];
    K.k[6 + 2 * i]     = y1[2 * i];
    K.k[6 + 2 * i + 1] = y1[2 * i + 1];
  }

  const int blocks = (B + 7) / 8;   // one wave32 per row, 8 waves per block
  hipLaunchKernelGGL(pt_sample_kernel, dim3(blocks), dim3(256), 0, stream,
                     order_mask, budgets, tile_masks, loop_ind, rbs, t2max,
                     mtt2, spmax, spmin, olog, tlog, splog, out, B, K);
}